// CenterLoss_67396626809078
// MI455X (gfx1250) — compile-verified
//
#include <hip/hip_runtime.h>
#include <hip/hip_bf16.h>

typedef __attribute__((ext_vector_type(2))) float v2f;
typedef __attribute__((ext_vector_type(8))) float v8f;

// Problem constants (from the reference)
constexpr int BATCH = 16384;
constexpr int FEAT  = 1024;
constexpr int WAVES_PER_BLOCK = 8;          // 256 threads, wave32
constexpr int BLOCK_THREADS   = 32 * WAVES_PER_BLOCK;
constexpr int TILES  = BATCH / 16;          // 1024 tiles of 16 rows
constexpr int BLOCKS = TILES / WAVES_PER_BLOCK;  // 128 blocks

// masked-out entries contribute batch*(num_classes-1)*1e-12, computed in f32
// exactly as the reference does (163823616 is exactly representable in f32).
#define MASKED_CONST (163823616.0f * 1e-12f)

// Stage 1: each wave computes squared distances for 16 rows via
// V_WMMA_F32_16X16X4_F32 with A == B == (pred - center) tile, so the
// diagonal of the accumulator is ||pred_i - c_i||^2 in exact f32 FMA math.
__global__ __launch_bounds__(BLOCK_THREADS)
void center_loss_stage1(const float* __restrict__ pred,
                        const float* __restrict__ centers,
                        const int*   __restrict__ target,
                        double*      __restrict__ partial) {
    const int lane   = threadIdx.x & 31;
    const int waveId = threadIdx.x >> 5;
    const int tile   = blockIdx.x * WAVES_PER_BLOCK + waveId;

    // f32 WMMA A/B lane mapping: lane L owns row (L&15), K-sub = 2*(L>>4)+{0,1}
    const int rowLocal = lane & 15;
    const int kSub     = (lane >> 4) * 2;
    const int row      = tile * 16 + rowLocal;

    const float* __restrict__ prow = pred    + (size_t)row * FEAT + kSub;
    const float* __restrict__ crow = centers + (size_t)target[row] * FEAT + kSub;

    v8f acc0 = {};
    v8f acc1 = {};

    // K loop: two independent 16x16x4 WMMA accumulations per step for ILP.
    #pragma unroll 2
    for (int k = 0; k < FEAT; k += 8) {
        v2f p0 = *(const v2f*)(prow + k);
        v2f c0 = *(const v2f*)(crow + k);
        v2f p1 = *(const v2f*)(prow + k + 4);
        v2f c1 = *(const v2f*)(crow + k + 4);
        v2f d0 = p0 - c0;
        v2f d1 = p1 - c1;
        // D = d * d^T + C : same register serves as A (16x4) and B (4x16)
        acc0 = __builtin_amdgcn_wmma_f32_16x16x4_f32(
            false, d0, false, d0, (short)0, acc0, false, false);
        acc1 = __builtin_amdgcn_wmma_f32_16x16x4_f32(
            false, d1, false, d1, (short)0, acc1, false, false);
    }

    // Diagonal extraction per the 16x16 f32 C/D layout:
    //   lanes 0..7   hold D[m][m], m = lane,      at component lane
    //   lanes 24..31 hold D[m][m], m = lane - 16, at component lane - 24
    float dist = 0.0f;
    if (lane < 8) {
        dist = acc0[lane] + acc1[lane];
    } else if (lane >= 24) {
        int comp = lane - 24;
        dist = acc0[comp] + acc1[comp];
    }
    if (lane < 8 || lane >= 24) {
        dist = fminf(fmaxf(dist, 1e-12f), 1e12f);   // reference clamp
    } else {
        dist = 0.0f;
    }

    // Deterministic fixed-order block reduction.
    __shared__ float red[BLOCK_THREADS];
    red[threadIdx.x] = dist;
    __syncthreads();
    #pragma unroll
    for (int off = BLOCK_THREADS / 2; off > 0; off >>= 1) {
        if (threadIdx.x < off) red[threadIdx.x] += red[threadIdx.x + off];
        __syncthreads();
    }
    if (threadIdx.x == 0) partial[blockIdx.x] = (double)red[0];
}

// Stage 2: reduce the 128 per-block partials (double), add masked constant.
__global__ __launch_bounds__(128)
void center_loss_stage2(const double* __restrict__ partial,
                        float* __restrict__ out) {
    __shared__ double s[128];
    const int t = threadIdx.x;
    s[t] = partial[t];
    __syncthreads();
    #pragma unroll
    for (int off = 64; off > 0; off >>= 1) {
        if (t < off) s[t] += s[t + off];
        __syncthreads();
    }
    if (t == 0) out[0] = (float)s[0] + MASKED_CONST;
}

extern "C" void kernel_launch(void* const* d_in, const int* in_sizes, int n_in,
                              void* d_out, int out_size, void* d_ws, size_t ws_size,
                              hipStream_t stream) {
    const float* pred    = (const float*)d_in[0];   // [16384, 1024] f32
    const float* centers = (const float*)d_in[1];   // [10000, 1024] f32
    const int*   target  = (const int*)d_in[2];     // [16384] int
    float*  out     = (float*)d_out;                // scalar f32
    double* partial = (double*)d_ws;                // 128 doubles of scratch

    center_loss_stage1<<<BLOCKS, BLOCK_THREADS, 0, stream>>>(pred, centers, target, partial);
    center_loss_stage2<<<1, 128, 0, stream>>>(partial, out);
}